// CIDER_42898133352759
// MI455X (gfx1250) — compile-verified
//
#include <hip/hip_runtime.h>
#include <hip/hip_bf16.h>
#include <math.h>

typedef __attribute__((ext_vector_type(2))) float v2f;
typedef __attribute__((ext_vector_type(8))) float v8f;

#define DEV_ATOMIC_ADD(p, v) \
    __hip_atomic_fetch_add((p), (v), __ATOMIC_RELAXED, __HIP_MEMORY_SCOPE_AGENT)

// ---------------- elementwise / degree kernels ----------------

__global__ void k_init_deg(float* deg, int n) {
    int i = blockIdx.x * blockDim.x + threadIdx.x;
    if (i < n) deg[i] = 1.0f;                      // self loop
}

__global__ void k_count_deg(const int* __restrict__ dst, float* deg, int E) {
    int i = blockIdx.x * blockDim.x + threadIdx.x;
    if (i < E) DEV_ATOMIC_ADD(&deg[dst[i]], 1.0f);
}

__global__ void k_rsqrt(float* d, int n) {
    int i = blockIdx.x * blockDim.x + threadIdx.x;
    if (i < n) d[i] = rsqrtf(d[i]);                // deg >= 1 always
}

// out[v,f] = bias[f] + pre[v,f] * dis[v]^2   (self-loop message + bias)
__global__ void k_self_init(const float* __restrict__ pre, const float* __restrict__ bias,
                            const float* __restrict__ dis, float* __restrict__ out,
                            int n, int F) {
    long long i = (long long)blockIdx.x * blockDim.x + threadIdx.x;
    if (i >= (long long)n * F) return;
    int v = (int)(i / F);
    int f = (int)(i - (long long)v * F);
    float dv = dis[v];
    out[i] = bias[f] + pre[i] * dv * dv;
}

__global__ void k_relu(float* h, long long n) {
    long long i = (long long)blockIdx.x * blockDim.x + threadIdx.x;
    if (i < n) h[i] = fmaxf(h[i], 0.0f);
}

__global__ void k_reparam(const float* __restrict__ mu, const float* __restrict__ lv,
                          const float* __restrict__ eps, float* __restrict__ z, long long n) {
    long long i = (long long)blockIdx.x * blockDim.x + threadIdx.x;
    if (i < n) z[i] = mu[i] + eps[i] * expf(0.5f * lv[i]);
}

// ---------------- edge scatter: agg[dst] += msg[src] * norm ----------------
// One thread handles one edge x 4 consecutive features (float4 gather + 4 fp32 atomics).
template <int GSHIFT>  // F = 4 << GSHIFT features per row
__global__ void k_scatter(const int* __restrict__ src, const int* __restrict__ dst,
                          const float* __restrict__ dis, const float* __restrict__ msg,
                          float* __restrict__ agg, int E) {
    constexpr int F = 4 << GSHIFT;
    long long i = (long long)blockIdx.x * blockDim.x + threadIdx.x;
    if (i >= ((long long)E << GSHIFT)) return;
    int e = (int)(i >> GSHIFT);
    int g = ((int)i & ((1 << GSHIFT) - 1)) * 4;
    int s = src[e], d = dst[e];
    float nm = dis[s] * dis[d];
    float4 v = *(const float4*)(msg + (size_t)s * F + g);
    float* p = agg + (size_t)d * F + g;
    DEV_ATOMIC_ADD(p + 0, v.x * nm);
    DEV_ATOMIC_ADD(p + 1, v.y * nm);
    DEV_ATOMIC_ADD(p + 2, v.z * nm);
    DEV_ATOMIC_ADD(p + 3, v.w * nm);
}

// ---------------- WMMA fp32 GEMM: C[nrows x cols] = act(A[nrows x K] @ B[K x bcols] + bias)
// One wave32 computes a 16-row tile across NCT column-tiles of 16 using
// V_WMMA_F32_16X16X4_F32.  B is staged in LDS (zero-padded to NCT*16 cols).
template <int NCT, int ACT, int COLLIM>
__global__ void k_gemm_wmma(const float* __restrict__ A, const float* __restrict__ B,
                            const float* __restrict__ bias, float* __restrict__ C,
                            int nrows, int K, int bcols, int ldc) {
    constexpr int WC = NCT * 16;
    extern __shared__ float Bs[];
    for (int i = threadIdx.x; i < K * WC; i += blockDim.x) {
        int k = i / WC, c = i - k * WC;
        Bs[i] = (c < bcols) ? B[k * bcols + c] : 0.0f;
    }
    __syncthreads();

    int wave = threadIdx.x >> 5;
    int lane = threadIdx.x & 31;
    int tile = blockIdx.x * (blockDim.x >> 5) + wave;
    int m0 = tile * 16;
    if (m0 >= nrows) return;                 // wave-uniform exit (EXEC stays all-ones)

    int half = lane >> 4;                    // A/B fragment: lanes 16-31 hold K+2,K+3
    int l = lane & 15;

    v8f acc[NCT];
#pragma unroll
    for (int ct = 0; ct < NCT; ++ct) acc[ct] = (v8f){0, 0, 0, 0, 0, 0, 0, 0};

    const float* arow = A + (size_t)(m0 + l) * K;
    for (int k0 = 0; k0 < K; k0 += 4) {
        float2 af = *(const float2*)(arow + k0 + 2 * half);   // 8B-aligned
        v2f a;
        a.x = af.x;
        a.y = af.y;
#pragma unroll
        for (int ct = 0; ct < NCT; ++ct) {
            v2f b;
            b.x = Bs[(k0 + 2 * half) * WC + ct * 16 + l];
            b.y = Bs[(k0 + 2 * half + 1) * WC + ct * 16 + l];
            acc[ct] = __builtin_amdgcn_wmma_f32_16x16x4_f32(
                false, a, false, b, (short)0, acc[ct], false, false);
        }
    }

#pragma unroll
    for (int ct = 0; ct < NCT; ++ct) {
        int col = ct * 16 + l;
        float bv = bias ? bias[col < bcols ? col : 0] : 0.0f;
#pragma unroll
        for (int v = 0; v < 8; ++v) {
            int row = m0 + v + 8 * half;     // C layout: VGPR v -> M = v + 8*half
            float val = acc[ct][v] + bv;
            if (ACT) val = tanhf(val);
            if (col < COLLIM) C[(size_t)row * ldc + col] = val;
        }
    }
}

// ---------------- decoder edge head: sigmoid(dot10(t2[e0], t2[e1])) ----------------
__global__ void k_edgedot(const int* __restrict__ src, const int* __restrict__ dst,
                          const float* __restrict__ t2, float* __restrict__ ew, int E) {
    int e = blockIdx.x * blockDim.x + threadIdx.x;
    if (e >= E) return;
    const float* pa = t2 + (size_t)src[e] * 10;
    const float* pb = t2 + (size_t)dst[e] * 10;
    float s = 0.0f;
#pragma unroll
    for (int f = 0; f < 10; ++f) s += pa[f] * pb[f];
    ew[e] = 1.0f / (1.0f + expf(-s));
}

// ---------------- host-side orchestration ----------------

static inline int cdiv(long long a, long long b) { return (int)((a + b - 1) / b); }

extern "C" void kernel_launch(void* const* d_in, const int* in_sizes, int n_in,
                              void* d_out, int out_size, void* d_ws, size_t ws_size,
                              hipStream_t stream) {
    const float* x        = (const float*)d_in[0];
    const int*   ei       = (const int*)d_in[1];
    const float* eps_c    = (const float*)d_in[2];
    const float* eps_n    = (const float*)d_in[3];
    const float* W_shared = (const float*)d_in[4];
    const float* b_shared = (const float*)d_in[5];
    const float* Wl[4] = {(const float*)d_in[6], (const float*)d_in[8],
                          (const float*)d_in[10], (const float*)d_in[12]};  // mu_c, mu_n, lv_c, lv_n
    const float* bl[4] = {(const float*)d_in[7], (const float*)d_in[9],
                          (const float*)d_in[11], (const float*)d_in[13]};
    const float* Wd1[2] = {(const float*)d_in[14], (const float*)d_in[18]};
    const float* bd1[2] = {(const float*)d_in[15], (const float*)d_in[19]};
    const float* Wd2[2] = {(const float*)d_in[16], (const float*)d_in[20]};
    const float* bd2[2] = {(const float*)d_in[17], (const float*)d_in[21]};

    const int N = in_sizes[0] / 128;
    const int E = in_sizes[1] / 2;
    const int* srcp = ei;
    const int* dstp = ei + E;

    // workspace layout (floats): dis | xw(32) | h(32) | pre(64) | z(64) | t1(32) | t2(10)
    float* ws  = (float*)d_ws;
    float* dis = ws;
    float* xw  = ws + (size_t)N;
    float* h   = ws + (size_t)33 * N;
    float* pre = ws + (size_t)65 * N;
    float* z   = ws + (size_t)129 * N;
    float* t1  = ws + (size_t)193 * N;
    float* t2  = ws + (size_t)225 * N;

    // output layout: ew_c[E] | ew_n[E] | mu_c | mu_n | lv_c | lv_n (each N*64)
    float* out = (float*)d_out;
    float* ewp[2]  = {out, out + E};
    float* head[4] = {out + 2LL * E,                     // mu_c
                      out + 2LL * E + 1LL * N * 64,      // mu_n
                      out + 2LL * E + 2LL * N * 64,      // lv_c
                      out + 2LL * E + 3LL * N * 64};     // lv_n

    const int T = 256;
    const int gemmBlocks = cdiv(N / 16 + ((N % 16) ? 1 : 0), T / 32);

    // 1) symmetric GCN normalization
    k_init_deg<<<cdiv(N, T), T, 0, stream>>>(dis, N);
    k_count_deg<<<cdiv(E, T), T, 0, stream>>>(dstp, dis, E);
    k_rsqrt<<<cdiv(N, T), T, 0, stream>>>(dis, N);

    // 2) shared GCN: h = relu(segsum(norm * (x@W)[src]) + b)
    k_gemm_wmma<2, 0, 32><<<gemmBlocks, T, 128 * 32 * sizeof(float), stream>>>(
        x, W_shared, nullptr, xw, N, 128, 32, 32);
    k_self_init<<<cdiv((long long)N * 32, T), T, 0, stream>>>(xw, b_shared, dis, h, N, 32);
    k_scatter<3><<<cdiv((long long)E * 8, T), T, 0, stream>>>(srcp, dstp, dis, xw, h, E);
    k_relu<<<cdiv((long long)N * 32, T), T, 0, stream>>>(h, (long long)N * 32);

    // 3) four GCN heads (mu_c, mu_n, lv_c, lv_n), pre-buffer reused stream-ordered
    for (int l = 0; l < 4; ++l) {
        k_gemm_wmma<4, 0, 64><<<gemmBlocks, T, 32 * 64 * sizeof(float), stream>>>(
            h, Wl[l], nullptr, pre, N, 32, 64, 64);
        k_self_init<<<cdiv((long long)N * 64, T), T, 0, stream>>>(pre, bl[l], dis, head[l], N, 64);
        k_scatter<4><<<cdiv((long long)E * 16, T), T, 0, stream>>>(srcp, dstp, dis, pre, head[l], E);
    }

    // 4) two decoders: z = mu + eps*exp(0.5*lv); t1 = tanh(z@Wd1+b); t2 = tanh(t1@Wd2+b);
    //    ew = sigmoid(dot10(t2[e0], t2[e1]))
    const float* eps[2] = {eps_c, eps_n};
    for (int dgt = 0; dgt < 2; ++dgt) {
        const float* mu = head[dgt];        // mu_c / mu_n
        const float* lv = head[2 + dgt];    // lv_c / lv_n
        k_reparam<<<cdiv((long long)N * 64, T), T, 0, stream>>>(mu, lv, eps[dgt], z,
                                                                (long long)N * 64);
        k_gemm_wmma<2, 1, 32><<<gemmBlocks, T, 64 * 32 * sizeof(float), stream>>>(
            z, Wd1[dgt], bd1[dgt], t1, N, 64, 32, 32);
        k_gemm_wmma<1, 1, 10><<<gemmBlocks, T, 32 * 16 * sizeof(float), stream>>>(
            t1, Wd2[dgt], bd2[dgt], t2, N, 32, 10, 10);
        k_edgedot<<<cdiv(E, T), T, 0, stream>>>(srcp, dstp, t2, ewp[dgt], E);
    }
}